// GNNSat_NESY_18940805776107
// MI455X (gfx1250) — compile-verified
//
#include <hip/hip_runtime.h>
#include <math.h>

#define NNODES 500000
#define NEDGES 16000000
#define EIROW  (2*NEDGES)        // length of one edge_index row
#define NB_RED 1024
#define RED_T  256
#define INV_SQRT2 0.70710678118654752440f
#define EPS_LN 1e-5

typedef float v2f __attribute__((ext_vector_type(2)));
typedef float v4f __attribute__((ext_vector_type(4)));
typedef float v8f __attribute__((ext_vector_type(8)));
typedef int   v2i __attribute__((ext_vector_type(2)));

// ---------- helpers ----------------------------------------------------------

// Exact float atomic max: positive-pattern values compare like ints,
// negative-pattern values compare reversed as uints. Init memory to 0xFF800000 (-inf).
__device__ __forceinline__ void atomicMaxF32(float* addr, float v) {
  unsigned bits = __float_as_uint(v);
  if ((bits >> 31) == 0) {
    atomicMax((int*)addr, (int)bits);           // +0.0 and positives
  } else {
    atomicMin((unsigned int*)addr, bits);       // -0.0 and negatives
  }
}

__device__ __forceinline__ double blockReduceSumD(double v) {
  __shared__ double sm[RED_T];
  int tid = threadIdx.x;
  sm[tid] = v;
  __syncthreads();
  for (int off = RED_T / 2; off > 0; off >>= 1) {
    if (tid < off) sm[tid] += sm[tid + off];
    __syncthreads();
  }
  double r = sm[0];
  __syncthreads();
  return r;
}

__device__ __forceinline__ float blockReduceMaxF(float v) {
  __shared__ float sm[RED_T];
  int tid = threadIdx.x;
  sm[tid] = v;
  __syncthreads();
  for (int off = RED_T / 2; off > 0; off >>= 1) {
    if (tid < off) sm[tid] = fmaxf(sm[tid], sm[tid + off]);
    __syncthreads();
  }
  float r = sm[0];
  __syncthreads();
  return r;
}

// ---------- node init --------------------------------------------------------

__global__ void init_nodes_kernel(unsigned* __restrict__ m, float* __restrict__ s,
                                  int* __restrict__ deg, unsigned* __restrict__ nmax) {
  int i = blockIdx.x * blockDim.x + threadIdx.x;
  if (i < NNODES) {
    m[i]    = 0xFF800000u;  // -inf
    s[i]    = 0.0f;
    deg[i]  = 0;
    nmax[i] = 0xFF800000u;  // -inf
  }
}

// ---------- node projections via V_WMMA_F32_16X16X4_F32 ----------------------
// One wave computes, for a tile of 16 nodes, D = A*B + C where
//   A (16x4 f32): row m = [x0, x1, 0, 0]        (lanes 0-15: K0/K1, lanes16-31: K2/K3 = 0)
//   B (4x16 f32): col n = row (n&1) of {Wq,Wk,Wv,Wskip}[n>>1] for n<8, else 0
//   C (16x16)   : C[m][n] = bias[n]
// so D[m][n] enumerates {q0,q1,k0,k1,v0,v1,skip0,skip1} for node m, exactly in f32.
__global__ void proj_kernel(const float* __restrict__ x,
                            const float* __restrict__ Wq, const float* __restrict__ bq,
                            const float* __restrict__ Wk, const float* __restrict__ bk,
                            const float* __restrict__ Wv, const float* __restrict__ bv,
                            const float* __restrict__ Wskip, const float* __restrict__ bskip,
                            float* __restrict__ proj) {
  const int lane = threadIdx.x & 31;
  const int wave = threadIdx.x >> 5;
  const int tile = blockIdx.x * (blockDim.x >> 5) + wave;
  const int ntiles = NNODES / 16;
  if (tile >= ntiles) return;                    // wave-uniform: EXEC stays all-ones

  const int n = lane & 15;
  float b0 = 0.f, b1 = 0.f, cb = 0.f;
  if (n < 8) {
    const float* Wsel = (n < 2) ? Wq : (n < 4) ? Wk : (n < 6) ? Wv : Wskip;
    const float* bsel = (n < 2) ? bq : (n < 4) ? bk : (n < 6) ? bv : bskip;
    int r = n & 1;
    if (lane < 16) { b0 = Wsel[2 * r]; b1 = Wsel[2 * r + 1]; }  // B rows 0,1 (rows 2,3 zero)
    cb = bsel[r];
  }

#if __has_builtin(__builtin_amdgcn_wmma_f32_16x16x4_f32)
  v2f a; a.x = 0.f; a.y = 0.f;
  if (lane < 16) {
    int node = tile * 16 + lane;
    a.x = x[2 * node];
    a.y = x[2 * node + 1];
  }
  v2f b; b.x = b0; b.y = b1;
  v8f c;
#pragma unroll
  for (int r = 0; r < 8; ++r) c[r] = cb;
  v8f d = __builtin_amdgcn_wmma_f32_16x16x4_f32(false, a, false, b, (short)0, c, false, false);

  // D layout: VGPR r, lanes0-15 -> M=r, lanes16-31 -> M=r+8, N = lane&15
  const int mbase = tile * 16 + ((lane >= 16) ? 8 : 0);
  if (n < 8) {
#pragma unroll
    for (int r = 0; r < 8; ++r) {
      proj[(mbase + r) * 8 + n] = d[r];
    }
  }
#else
  // scalar fallback (kept so the file always compiles)
  if (lane < 16) {
    int node = tile * 16 + lane;
    float x0 = x[2 * node], x1 = x[2 * node + 1];
    float* p = proj + node * 8;
    p[0] = x0 * Wq[0] + x1 * Wq[1] + bq[0];
    p[1] = x0 * Wq[2] + x1 * Wq[3] + bq[1];
    p[2] = x0 * Wk[0] + x1 * Wk[1] + bk[0];
    p[3] = x0 * Wk[2] + x1 * Wk[3] + bk[1];
    p[4] = x0 * Wv[0] + x1 * Wv[1] + bv[0];
    p[5] = x0 * Wv[2] + x1 * Wv[3] + bv[1];
    p[6] = x0 * Wskip[0] + x1 * Wskip[1] + bskip[0];
    p[7] = x0 * Wskip[2] + x1 * Wskip[3] + bskip[1];
  }
  (void)b0; (void)b1; (void)cb;
#endif
}

// ---------- edge pass 1: alpha + segment max + degree ------------------------
// MODE 0: no side outputs; MODE 1: cache alpha; MODE 2: also pack (src,dst) and e.

template <int MODE>
__global__ void edge_alpha_kernel(const int* __restrict__ ei, const float* __restrict__ eattr,
                                  const float* __restrict__ We, const float* __restrict__ proj,
                                  float* __restrict__ m, int* __restrict__ deg,
                                  float* __restrict__ alpha, v2i* __restrict__ pairs,
                                  v2f* __restrict__ epack) {
  int i = blockIdx.x * blockDim.x + threadIdx.x;
  if (i >= NEDGES) return;
  v2i sv = __builtin_nontemporal_load(((const v2i*)ei) + i);                   // [2i,2i+1] row0
  v2i dv = __builtin_nontemporal_load(((const v2i*)(ei + (size_t)EIROW)) + i); // row1
  int src = sv.y;
  int dst = dv.y;
  v4f e4 = __builtin_nontemporal_load(((const v4f*)eattr) + i);  // rows 2i (clause), 2i+1 (attn)
  float e0 = e4.z * We[0] + e4.w * We[1];
  float e1 = e4.z * We[2] + e4.w * We[3];
  const float* pd = proj + (size_t)dst * 8;
  const float* ps = proj + (size_t)src * 8;
  float al = (pd[0] * (ps[2] + e0) + pd[1] * (ps[3] + e1)) * INV_SQRT2;
  atomicMaxF32(&m[dst], al);
  atomicAdd(&deg[dst], 1);
  if (MODE >= 1) alpha[i] = al;
  if (MODE == 2) {
    v2i p; p.x = src; p.y = dst;
    pairs[i] = p;
    v2f e2; e2.x = e0; e2.y = e1;
    epack[i] = e2;
  }
}

// fix m (-inf -> 0) and seed h with the skip projection
__global__ void fix_m_seed_h_kernel(float* __restrict__ m, const float* __restrict__ proj,
                                    float* __restrict__ h) {
  int i = blockIdx.x * blockDim.x + threadIdx.x;
  if (i < NNODES) {
    if (__float_as_uint(m[i]) == 0xFF800000u) m[i] = 0.0f;  // where(isfinite(m), m, 0)
    h[2 * i + 0] = proj[(size_t)i * 8 + 6];
    h[2 * i + 1] = proj[(size_t)i * 8 + 7];
  }
}

// ---------- edge pass 2: a = exp(alpha - m[dst]); s[dst] += a ----------------

template <int MODE>
__global__ void edge_expsum_kernel(const int* __restrict__ ei, const float* __restrict__ eattr,
                                   const float* __restrict__ We, const float* __restrict__ proj,
                                   const float* __restrict__ m, float* __restrict__ s,
                                   float* __restrict__ alpha, const v2i* __restrict__ pairs) {
  int i = blockIdx.x * blockDim.x + threadIdx.x;
  if (i >= NEDGES) return;
  int dst;
  float al;
  if (MODE == 2) {
    v2i p = pairs[i];
    dst = p.y;
    al = alpha[i];
  } else if (MODE == 1) {
    v2i dv = __builtin_nontemporal_load(((const v2i*)(ei + (size_t)EIROW)) + i);
    dst = dv.y;
    al = alpha[i];
  } else {
    v2i sv = __builtin_nontemporal_load(((const v2i*)ei) + i);
    v2i dv = __builtin_nontemporal_load(((const v2i*)(ei + (size_t)EIROW)) + i);
    dst = dv.y;
    int src = sv.y;
    v4f e4 = __builtin_nontemporal_load(((const v4f*)eattr) + i);
    float e0 = e4.z * We[0] + e4.w * We[1];
    float e1 = e4.z * We[2] + e4.w * We[3];
    const float* pd = proj + (size_t)dst * 8;
    const float* ps = proj + (size_t)src * 8;
    al = (pd[0] * (ps[2] + e0) + pd[1] * (ps[3] + e1)) * INV_SQRT2;
  }
  float a = expf(al - m[dst]);
  if (MODE >= 1) alpha[i] = a;           // overwrite: pass 3 reads exp value directly
  atomicAdd(&s[dst], a);
}

// ---------- edge pass 3: weighted message scatter ----------------------------

template <int MODE>
__global__ void edge_scatter_kernel(const int* __restrict__ ei, const float* __restrict__ eattr,
                                    const float* __restrict__ We, const float* __restrict__ proj,
                                    const float* __restrict__ m, const float* __restrict__ s,
                                    const float* __restrict__ alpha, const v2i* __restrict__ pairs,
                                    const v2f* __restrict__ epack, float* __restrict__ h) {
  int i = blockIdx.x * blockDim.x + threadIdx.x;
  if (i >= NEDGES) return;
  int src, dst;
  float e0, e1, a;
  if (MODE == 2) {
    v2i p = pairs[i];
    src = p.x; dst = p.y;
    v2f e2 = epack[i];
    e0 = e2.x; e1 = e2.y;
    a = alpha[i];
  } else {
    v2i sv = __builtin_nontemporal_load(((const v2i*)ei) + i);
    v2i dv = __builtin_nontemporal_load(((const v2i*)(ei + (size_t)EIROW)) + i);
    src = sv.y; dst = dv.y;
    v4f e4 = __builtin_nontemporal_load(((const v4f*)eattr) + i);
    e0 = e4.z * We[0] + e4.w * We[1];
    e1 = e4.z * We[2] + e4.w * We[3];
    if (MODE == 1) {
      a = alpha[i];
    } else {
      const float* pd = proj + (size_t)dst * 8;
      const float* ps = proj + (size_t)src * 8;
      float al = (pd[0] * (ps[2] + e0) + pd[1] * (ps[3] + e1)) * INV_SQRT2;
      a = expf(al - m[dst]);
    }
  }
  float sd = s[dst];
  float w = a / (sd > 0.0f ? sd : 1.0f);
  const float* ps = proj + (size_t)src * 8;
  atomicAdd(&h[2 * dst + 0], (ps[4] + e0) * w);
  atomicAdd(&h[2 * dst + 1], (ps[5] + e1) * w);
}

// ---------- batch-norm stats (deterministic two-stage, double partials) ------

__global__ void stats_sum_kernel(const float* __restrict__ h, double* __restrict__ part) {
  double s0 = 0, s1 = 0, q0 = 0, q1 = 0;
  for (int i = blockIdx.x * blockDim.x + threadIdx.x; i < NNODES; i += gridDim.x * blockDim.x) {
    double a = h[2 * i], b = h[2 * i + 1];
    s0 += a; s1 += b; q0 += a * a; q1 += b * b;
  }
  s0 = blockReduceSumD(s0);
  s1 = blockReduceSumD(s1);
  q0 = blockReduceSumD(q0);
  q1 = blockReduceSumD(q1);
  if (threadIdx.x == 0) {
    part[blockIdx.x * 4 + 0] = s0;
    part[blockIdx.x * 4 + 1] = s1;
    part[blockIdx.x * 4 + 2] = q0;
    part[blockIdx.x * 4 + 3] = q1;
  }
}

__global__ void finalize_stats_kernel(const double* __restrict__ part, float* __restrict__ stats) {
  double s0 = 0, s1 = 0, q0 = 0, q1 = 0;
  for (int i = threadIdx.x; i < NB_RED; i += blockDim.x) {
    s0 += part[i * 4 + 0]; s1 += part[i * 4 + 1];
    q0 += part[i * 4 + 2]; q1 += part[i * 4 + 3];
  }
  s0 = blockReduceSumD(s0);
  s1 = blockReduceSumD(s1);
  q0 = blockReduceSumD(q0);
  q1 = blockReduceSumD(q1);
  if (threadIdx.x == 0) {
    double invN = 1.0 / (double)NNODES;
    double mu0 = s0 * invN, mu1 = s1 * invN;
    double var0 = q0 * invN - mu0 * mu0;
    double var1 = q1 * invN - mu1 * mu1;
    stats[0] = (float)mu0;
    stats[1] = (float)mu1;
    stats[2] = (float)(1.0 / sqrt(var0 + EPS_LN));
    stats[3] = (float)(1.0 / sqrt(var1 + EPS_LN));
  }
}

// ---------- normalize + column max -------------------------------------------

__global__ void norm_max_kernel(float* __restrict__ h, const float* __restrict__ stats,
                                const float* __restrict__ gamma, const float* __restrict__ beta,
                                float* __restrict__ pmax) {
  float mu0 = stats[0], mu1 = stats[1], rs0 = stats[2], rs1 = stats[3];
  float g0 = gamma[0], g1 = gamma[1], be0 = beta[0], be1 = beta[1];
  float mx0 = -__builtin_inff(), mx1 = -__builtin_inff();
  for (int i = blockIdx.x * blockDim.x + threadIdx.x; i < NNODES; i += gridDim.x * blockDim.x) {
    float a = (h[2 * i] - mu0) * rs0 * g0 + be0;
    float b = (h[2 * i + 1] - mu1) * rs1 * g1 + be1;
    h[2 * i] = a; h[2 * i + 1] = b;
    mx0 = fmaxf(mx0, a); mx1 = fmaxf(mx1, b);
  }
  mx0 = blockReduceMaxF(mx0);
  mx1 = blockReduceMaxF(mx1);
  if (threadIdx.x == 0) { pmax[blockIdx.x * 2 + 0] = mx0; pmax[blockIdx.x * 2 + 1] = mx1; }
}

__global__ void finalize_max_kernel(const float* __restrict__ pmax, float* __restrict__ stats) {
  float mx0 = -__builtin_inff(), mx1 = -__builtin_inff();
  for (int i = threadIdx.x; i < NB_RED; i += blockDim.x) {
    mx0 = fmaxf(mx0, pmax[i * 2 + 0]);
    mx1 = fmaxf(mx1, pmax[i * 2 + 1]);
  }
  mx0 = blockReduceMaxF(mx0);
  mx1 = blockReduceMaxF(mx1);
  if (threadIdx.x == 0) { stats[4] = mx0; stats[5] = mx1; }
}

// ---------- softmax numerator + column sum -----------------------------------

__global__ void exp_colsum_kernel(float* __restrict__ h, const float* __restrict__ stats,
                                  double* __restrict__ part) {
  float c0 = stats[4], c1 = stats[5];
  double s0 = 0, s1 = 0;
  for (int i = blockIdx.x * blockDim.x + threadIdx.x; i < NNODES; i += gridDim.x * blockDim.x) {
    float p0 = expf(h[2 * i] - c0);
    float p1 = expf(h[2 * i + 1] - c1);
    h[2 * i] = p0; h[2 * i + 1] = p1;    // store unnormalized exp; divide lazily via stats[6..7]
    s0 += p0; s1 += p1;
  }
  s0 = blockReduceSumD(s0);
  s1 = blockReduceSumD(s1);
  if (threadIdx.x == 0) { part[blockIdx.x * 2 + 0] = s0; part[blockIdx.x * 2 + 1] = s1; }
}

__global__ void finalize_colsum_kernel(const double* __restrict__ part, float* __restrict__ stats) {
  double s0 = 0, s1 = 0;
  for (int i = threadIdx.x; i < NB_RED; i += blockDim.x) {
    s0 += part[i * 2 + 0]; s1 += part[i * 2 + 1];
  }
  s0 = blockReduceSumD(s0);
  s1 = blockReduceSumD(s1);
  if (threadIdx.x == 0) {
    stats[6] = (float)(1.0 / s0);
    stats[7] = (float)(1.0 / s1);
  }
}

// ---------- clause pass: neighbor max ----------------------------------------

template <bool PACKED>
__global__ void edge_clause_kernel(const int* __restrict__ ei, const float* __restrict__ eattr,
                                   const float* __restrict__ h, const float* __restrict__ stats,
                                   float* __restrict__ nmax, const v2i* __restrict__ pairs) {
  int i = blockIdx.x * blockDim.x + threadIdx.x;
  if (i >= NEDGES) return;
  int src, dst;
  if (PACKED) {
    v2i p = pairs[i];
    src = p.x; dst = p.y;
  } else {
    v2i sv = __builtin_nontemporal_load(((const v2i*)ei) + i);
    v2i dv = __builtin_nontemporal_load(((const v2i*)(ei + (size_t)EIROW)) + i);
    src = sv.y; dst = dv.y;
  }
  v4f e4 = __builtin_nontemporal_load(((const v4f*)eattr) + i);  // (x,y) = ea_clause row 2i
  float invS0 = stats[6], invS1 = stats[7];
  float sc = h[2 * src] * invS0 * e4.x + h[2 * src + 1] * invS1 * e4.y;
  atomicMaxF32(&nmax[dst], sc);
}

// ---------- final output ------------------------------------------------------

__global__ void final_kernel(const float* __restrict__ h, const float* __restrict__ stats,
                             const int* __restrict__ deg, const float* __restrict__ nmax,
                             const float* __restrict__ mask, const float* __restrict__ Wf,
                             const float* __restrict__ bf, float* __restrict__ out) {
  int i = blockIdx.x * blockDim.x + threadIdx.x;
  if (i >= NNODES) return;
  float invS0 = stats[6], invS1 = stats[7];
  float hf0, hf1;
  if (deg[i] > 0) {
    float mx = nmax[i];
    hf0 = mx;
    hf1 = 1.0f - mx;
  } else {
    hf0 = h[2 * i] * invS0;        // softmaxed value
    hf1 = h[2 * i + 1] * invS1;
  }
  out[i] = (hf0 * Wf[0] + hf1 * Wf[1] + bf[0]) * mask[i];
}

// ---------- host launch -------------------------------------------------------

extern "C" void kernel_launch(void* const* d_in, const int* in_sizes, int n_in,
                              void* d_out, int out_size, void* d_ws, size_t ws_size,
                              hipStream_t stream) {
  (void)in_sizes; (void)n_in; (void)out_size;
  const float* x     = (const float*)d_in[0];
  const int*   ei    = (const int*)d_in[1];
  const float* eattr = (const float*)d_in[2];
  const float* mask  = (const float*)d_in[3];
  const float* Wq    = (const float*)d_in[4];
  const float* bq    = (const float*)d_in[5];
  const float* Wk    = (const float*)d_in[6];
  const float* bk    = (const float*)d_in[7];
  const float* Wv    = (const float*)d_in[8];
  const float* bv    = (const float*)d_in[9];
  const float* We    = (const float*)d_in[10];
  const float* Wskip = (const float*)d_in[11];
  const float* bskip = (const float*)d_in[12];
  const float* gamma = (const float*)d_in[13];
  const float* beta  = (const float*)d_in[14];
  const float* Wf    = (const float*)d_in[15];
  const float* bf    = (const float*)d_in[16];
  float* out = (float*)d_out;

  char* base = (char*)d_ws;
  size_t off = 0;
  auto carve = [&](size_t bytes) -> char* {
    char* p = base + off;
    off += (bytes + 255) & ~(size_t)255;
    return p;
  };
  float*  proj  = (float*)carve((size_t)NNODES * 8 * sizeof(float));
  float*  h     = (float*)carve((size_t)NNODES * 2 * sizeof(float));
  float*  mseg  = (float*)carve((size_t)NNODES * sizeof(float));
  float*  sseg  = (float*)carve((size_t)NNODES * sizeof(float));
  int*    deg   = (int*)  carve((size_t)NNODES * sizeof(int));
  float*  nmax  = (float*)carve((size_t)NNODES * sizeof(float));
  double* part  = (double*)carve((size_t)NB_RED * 4 * sizeof(double));
  float*  pmax  = (float*)carve((size_t)NB_RED * 2 * sizeof(float));
  float*  stats = (float*)carve(64);
  // optional tiers
  float*  alpha = (float*)carve((size_t)NEDGES * sizeof(float));   // tier >= 1
  size_t need1 = off;
  v2i*    pairs = (v2i*)carve((size_t)NEDGES * sizeof(v2i));       // tier 2
  v2f*    epack = (v2f*)carve((size_t)NEDGES * sizeof(v2f));       // tier 2
  size_t need2 = off;
  int mode = (ws_size >= need2) ? 2 : (ws_size >= need1) ? 1 : 0;

  const int tb = 256;
  const int nodeBlocks = (NNODES + tb - 1) / tb;
  const int edgeBlocks = (NEDGES + tb - 1) / tb;
  const int ntiles = NNODES / 16;
  const int projBlocks = (ntiles + (tb / 32) - 1) / (tb / 32);

  init_nodes_kernel<<<nodeBlocks, tb, 0, stream>>>((unsigned*)mseg, sseg, deg, (unsigned*)nmax);
  proj_kernel<<<projBlocks, tb, 0, stream>>>(x, Wq, bq, Wk, bk, Wv, bv, Wskip, bskip, proj);

  if (mode == 2) {
    edge_alpha_kernel<2><<<edgeBlocks, tb, 0, stream>>>(ei, eattr, We, proj, mseg, deg, alpha, pairs, epack);
  } else if (mode == 1) {
    edge_alpha_kernel<1><<<edgeBlocks, tb, 0, stream>>>(ei, eattr, We, proj, mseg, deg, alpha, pairs, epack);
  } else {
    edge_alpha_kernel<0><<<edgeBlocks, tb, 0, stream>>>(ei, eattr, We, proj, mseg, deg, alpha, pairs, epack);
  }

  fix_m_seed_h_kernel<<<nodeBlocks, tb, 0, stream>>>(mseg, proj, h);

  if (mode == 2) {
    edge_expsum_kernel<2><<<edgeBlocks, tb, 0, stream>>>(ei, eattr, We, proj, mseg, sseg, alpha, pairs);
    edge_scatter_kernel<2><<<edgeBlocks, tb, 0, stream>>>(ei, eattr, We, proj, mseg, sseg, alpha, pairs, epack, h);
  } else if (mode == 1) {
    edge_expsum_kernel<1><<<edgeBlocks, tb, 0, stream>>>(ei, eattr, We, proj, mseg, sseg, alpha, pairs);
    edge_scatter_kernel<1><<<edgeBlocks, tb, 0, stream>>>(ei, eattr, We, proj, mseg, sseg, alpha, pairs, epack, h);
  } else {
    edge_expsum_kernel<0><<<edgeBlocks, tb, 0, stream>>>(ei, eattr, We, proj, mseg, sseg, alpha, pairs);
    edge_scatter_kernel<0><<<edgeBlocks, tb, 0, stream>>>(ei, eattr, We, proj, mseg, sseg, alpha, pairs, epack, h);
  }

  stats_sum_kernel<<<NB_RED, RED_T, 0, stream>>>(h, part);
  finalize_stats_kernel<<<1, RED_T, 0, stream>>>(part, stats);
  norm_max_kernel<<<NB_RED, RED_T, 0, stream>>>(h, stats, gamma, beta, pmax);
  finalize_max_kernel<<<1, RED_T, 0, stream>>>(pmax, stats);
  exp_colsum_kernel<<<NB_RED, RED_T, 0, stream>>>(h, stats, part);
  finalize_colsum_kernel<<<1, RED_T, 0, stream>>>(part, stats);

  if (mode == 2) {
    edge_clause_kernel<true><<<edgeBlocks, tb, 0, stream>>>(ei, eattr, h, stats, nmax, pairs);
  } else {
    edge_clause_kernel<false><<<edgeBlocks, tb, 0, stream>>>(ei, eattr, h, stats, nmax, pairs);
  }
  final_kernel<<<nodeBlocks, tb, 0, stream>>>(h, stats, deg, nmax, mask, Wf, bf, out);
}